// VolumeRenderer_46256797778604
// MI455X (gfx1250) — compile-verified
//
#include <hip/hip_runtime.h>

// Volume rendering (NeRF-style) for MI455X / gfx1250.
// Bandwidth-bound: 640MB read -> ~27.5us floor @ 23.3 TB/s.
// Strategy: wave-per-16-rays, coalesced float4 loads, cumsum-as-triangular-
// matmul on the WMMA pipe (v_wmma_f32_16x16x4_f32), wave32 shuffle reductions.

typedef float v2f __attribute__((ext_vector_type(2)));
typedef float v8f __attribute__((ext_vector_type(8)));

#define S_SAMPLES 128
#define RAYS_PER_WAVE 16
#define WAVES_PER_BLOCK 2
#define BLOCK_THREADS (32 * WAVES_PER_BLOCK)
#define RAYS_PER_BLOCK (RAYS_PER_WAVE * WAVES_PER_BLOCK)
#define USTRIDE 132          // 128 + 4 pad: row-stride LDS reads hit distinct banks
#define FAR_DELTA 1e10f

__global__ __launch_bounds__(BLOCK_THREADS)
void VolumeRenderer_gfx1250_kernel(const float* __restrict__ density,
                                   const float* __restrict__ feature,
                                   const float* __restrict__ depth,
                                   float* __restrict__ out,
                                   int nRays)
{
    __shared__ float U[WAVES_PER_BLOCK][RAYS_PER_WAVE * USTRIDE];   // u terms
    __shared__ float Aw[WAVES_PER_BLOCK][RAYS_PER_WAVE * USTRIDE];  // 1-exp(-d*delta)

    const int lane = threadIdx.x & 31;
    const int wave = threadIdx.x >> 5;
    const int rayBase = blockIdx.x * RAYS_PER_BLOCK + wave * RAYS_PER_WAVE;
    if (rayBase + RAYS_PER_WAVE > nRays) return;   // wave-uniform: EXEC stays full

    float* __restrict__ Uw  = U[wave];
    float* __restrict__ Aww = Aw[wave];

    // ---------------- Phase A: coalesced load + per-sample terms -> LDS ------
    // Lane l owns samples 4l..4l+3 of ray t. Boundary depths via wave shuffles.
    const int s0 = lane * 4;
    for (int t = 0; t < RAYS_PER_WAVE; ++t) {
        const size_t base = (size_t)(rayBase + t) * S_SAMPLES + s0;
        float4 dp = *(const float4*)(depth   + base);   // global_load_b128
        float4 dn = *(const float4*)(density + base);   // global_load_b128

        float prevw = __shfl_up(dp.w, 1, 32);           // depth[4l-1]
        float nextx = __shfl_down(dp.x, 1, 32);         // depth[4l+4]

        // u[s] = d[s]*(depth[s]-depth[s-1]); u[0] = 0
        float u0 = (lane == 0) ? 0.0f : dn.x * (dp.x - prevw);
        float u1 = dn.y * (dp.y - dp.x);
        float u2 = dn.z * (dp.z - dp.y);
        float u3 = dn.w * (dp.w - dp.z);

        // a[s] = 1 - exp(-d[s]*delta[s]); delta[S-1] = 1e10 sentinel
        float dl3 = (lane == 31) ? FAR_DELTA : (nextx - dp.w);
        float a0 = 1.0f - __expf(-dn.x * (dp.y - dp.x));
        float a1 = 1.0f - __expf(-dn.y * (dp.z - dp.y));
        float a2 = 1.0f - __expf(-dn.z * (dp.w - dp.z));
        float a3 = 1.0f - __expf(-dn.w * dl3);

        *(float4*)(Uw  + t * USTRIDE + s0) = make_float4(u0, u1, u2, u3);
        *(float4*)(Aww + t * USTRIDE + s0) = make_float4(a0, a1, a2, a3);
    }
    // Wave-private LDS slab: no cross-wave sharing, dscnt ordering suffices.

    // ---------------- Phase B: cumsum via triangular WMMA + fused epilogue ---
    // cs[r][j] = sum_{k<=j} u[r][k]  ==  U_row x UpperTriOnes.
    // Per 16-col tile jb:  cs_tile = A_jb x Tri + running ; running += A_jb x Ones.
    const int j    = lane & 15;     // column (sample within tile) / A row
    const int half = lane >> 4;
    const int koff = half * 2;      // A/B VGPR0 holds K={0,2}, VGPR1 K={1,3}
    const int mc   = half << 3;     // C/D rows: vgpr i -> ray i + 8*half

    const v2f onesB = {1.0f, 1.0f};
    v8f run = {};                   // running row-sums, broadcast across columns
    float acc[8][3];
    #pragma unroll
    for (int i = 0; i < 8; ++i) { acc[i][0] = acc[i][1] = acc[i][2] = 0.0f; }

    for (int jb = 0; jb < 8; ++jb) {
        // A slices for this 16-sample tile (K=4 each), WMMA A layout:
        // lane m(0-15): A[m][4q+0..1] ; lane m+16: A[m][4q+2..3]
        v2f Asl[4];
        #pragma unroll
        for (int q = 0; q < 4; ++q)
            Asl[q] = *(const v2f*)(Uw + j * USTRIDE + jb * 16 + q * 4 + koff);

        v8f cs = run;
        #pragma unroll
        for (int q = 0; q < 4; ++q) {
            int kk = q * 4 + koff;                     // global-in-tile K row
            v2f triB;
            triB.x = (kk     <= j) ? 1.0f : 0.0f;      // B[k][j] = (k <= j)
            triB.y = (kk + 1 <= j) ? 1.0f : 0.0f;
            cs = __builtin_amdgcn_wmma_f32_16x16x4_f32(
                     false, Asl[q], false, triB, (short)0, cs, false, false);
        }
        #pragma unroll
        for (int q = 0; q < 4; ++q) {
            run = __builtin_amdgcn_wmma_f32_16x16x4_f32(
                      false, Asl[q], false, onesB, (short)0, run, false, false);
        }

        // Epilogue on the D tile: element (vgpr i, lane) = (ray mc+i, sample jb*16+j)
        #pragma unroll
        for (int i = 0; i < 8; ++i) {
            const int mrow = mc + i;
            const int sc   = jb * 16 + j;
            float T = __expf(-cs[i]);                       // v_exp_f32
            float w = T * Aww[mrow * USTRIDE + sc];
            const float* f =
                feature + ((size_t)(rayBase + mrow) * S_SAMPLES + sc) * 3;
            acc[i][0] += w * f[0];
            acc[i][1] += w * f[1];
            acc[i][2] += w * f[2];
        }
    }

    // ---------------- Reduce 16 columns per half-wave, store [N,3] ----------
    #pragma unroll
    for (int i = 0; i < 8; ++i) {
        #pragma unroll
        for (int c = 0; c < 3; ++c) {
            float v = acc[i][c];
            v += __shfl_xor(v, 1, 32);
            v += __shfl_xor(v, 2, 32);
            v += __shfl_xor(v, 4, 32);
            v += __shfl_xor(v, 8, 32);   // stays within each 16-lane half
            acc[i][c] = v;
        }
    }
    if (j == 0) {                        // lanes 0 and 16 write 8 rays each
        #pragma unroll
        for (int i = 0; i < 8; ++i) {
            const size_t ray = (size_t)(rayBase + mc + i);
            out[ray * 3 + 0] = acc[i][0];
            out[ray * 3 + 1] = acc[i][1];
            out[ray * 3 + 2] = acc[i][2];
        }
    }
}

extern "C" void kernel_launch(void* const* d_in, const int* in_sizes, int n_in,
                              void* d_out, int out_size, void* d_ws, size_t ws_size,
                              hipStream_t stream) {
    const float* density = (const float*)d_in[0];   // [N, S, 1]
    const float* feature = (const float*)d_in[1];   // [N, S, 3]
    const float* depth   = (const float*)d_in[2];   // [N, S]
    float* out = (float*)d_out;                     // [N, 3]

    const int nRays = in_sizes[2] / S_SAMPLES;
    const int grid  = (nRays + RAYS_PER_BLOCK - 1) / RAYS_PER_BLOCK;

    VolumeRenderer_gfx1250_kernel<<<grid, BLOCK_THREADS, 0, stream>>>(
        density, feature, depth, out, nRays);
}